// Loss_58042188038797
// MI455X (gfx1250) — compile-verified
//
#include <hip/hip_runtime.h>

typedef __attribute__((ext_vector_type(2))) float v2f;
typedef __attribute__((ext_vector_type(4))) float v4f;
typedef __attribute__((ext_vector_type(8))) float v8f;
typedef __attribute__((ext_vector_type(2))) int   v2i;

#define WAVE 32
#define BLOCK_THREADS 256
#define NUM_BLOCKS 2048
#define LN2F 0.69314718055994530942f

// per-element term: (ln(p_win) - (f0 - 0.5)^2) * r_i   (R=1, P1=0.5)
// ln(p) computed as v_log_f32 (log2) * ln2 — inputs are in (0.01, 0.99),
// so no denorm/inf fixups needed.
__device__ __forceinline__ float term(float fx, float fy, int victor, int e,
                                      float inv_denom) {
    float p  = (victor == 0) ? fx : fy;
    float l  = __builtin_amdgcn_logf(p) * LN2F;
    float d  = fx - 0.5f;
    float ri = (float)e * inv_denom;
    return (l - d * d) * ri;
}

// Kernel 1: per-wave streaming reduction through V_WMMA_F32_16X16X4_F32.
// Each wave consumes a 128-element tile per iteration (4 per lane, b128+b64
// loads), computes the per-element terms, and accumulates all of them into a
// 16x16 f32 WMMA accumulator via B = all-ones (D[m][n] = sum_k A[m][k] + C).
// Loop bounds are forced wave-uniform (readfirstlane) so the loop is a scalar
// s_cmp/s_cbranch loop and EXEC stays all-ones at every WMMA (ISA req).
__global__ void loss_partial_kernel(const float* __restrict__ final_out,
                                    const int* __restrict__ point_victor,
                                    float* __restrict__ partials,
                                    int n, float inv_denom) {
    const int lane            = threadIdx.x & (WAVE - 1);
    const int wave_in_block   = threadIdx.x >> 5;
    const int waves_per_block = blockDim.x >> 5;
    const int total_waves     = gridDim.x * waves_per_block;  // uniform
    const int n_tiles         = n >> 7;                       // 128 elems/tile
    // force the wave id into an SGPR -> uniform loop, EXEC untouched
    const int global_wave =
        __builtin_amdgcn_readfirstlane(blockIdx.x * waves_per_block + wave_in_block);

    const v4f* __restrict__ fo4 = (const v4f*)final_out;   // 2 elements / v4f
    const v2i* __restrict__ pv2 = (const v2i*)point_victor;

    v8f acc = {};                             // 16x16 f32 accumulator (8 VGPRs)
    v2f ones; ones.x = 1.0f; ones.y = 1.0f;   // B matrix: 4x16 of 1.0

    for (int tile = global_wave; tile < n_tiles; tile += total_waves) {
        const int base = tile << 7;
        const int q0   = (base >> 1) + lane;  // pair-index: elements 2q, 2q+1
        const int q1   = q0 + WAVE;

        v4f fA = __builtin_nontemporal_load(&fo4[q0]);   // elems 2q0, 2q0+1
        v4f fB = __builtin_nontemporal_load(&fo4[q1]);   // elems 2q1, 2q1+1
        v2i vA = __builtin_nontemporal_load(&pv2[q0]);
        v2i vB = __builtin_nontemporal_load(&pv2[q1]);

        const int eA = q0 << 1;
        const int eB = q1 << 1;

        v2f a0, a1;   // A fragments (element->slot mapping arbitrary; B==ones)
        a0.x = term(fA.x, fA.y, vA.x, eA,     inv_denom);
        a0.y = term(fA.z, fA.w, vA.y, eA + 1, inv_denom);
        a1.x = term(fB.x, fB.y, vB.x, eB,     inv_denom);
        a1.y = term(fB.z, fB.w, vB.y, eB + 1, inv_denom);

        acc = __builtin_amdgcn_wmma_f32_16x16x4_f32(
            false, a0, false, ones, (short)0, acc, false, false);
        acc = __builtin_amdgcn_wmma_f32_16x16x4_f32(
            false, a1, false, ones, (short)0, acc, false, false);
    }

    // C/D layout: VGPR r, lanes 0-15 hold D[r][lane] = S_r (row-sum, identical
    // across columns since B == ones); lanes 16-31 hold S_{r+8}. Per-lane sum
    // of the 8 VGPRs => lane0 carries S_0..S_7, lane16 carries S_8..S_15.
    float s = acc[0] + acc[1] + acc[2] + acc[3] + acc[4] + acc[5] + acc[6] + acc[7];
    int   si = __builtin_bit_cast(int, s);
    float wave_sum =
        __builtin_bit_cast(float, __builtin_amdgcn_readlane(si, 0)) +
        __builtin_bit_cast(float, __builtin_amdgcn_readlane(si, 16));

    __shared__ float wsum[BLOCK_THREADS / WAVE];
    if (lane == 0) wsum[wave_in_block] = wave_sum;
    __syncthreads();
    if (threadIdx.x == 0) {
        float b = 0.0f;
        for (int i = 0; i < waves_per_block; ++i) b += wsum[i];
        partials[blockIdx.x] = b;   // every block writes -> d_ws fully initialized
    }
}

// Kernel 2: deterministic single-block tree reduction of block partials,
// scalar tail handling (n = 2^24 here, so no tail), negate, write scalar.
__global__ void loss_final_kernel(const float* __restrict__ partials, int num_partials,
                                  const float* __restrict__ final_out,
                                  const int* __restrict__ point_victor,
                                  int n, float inv_denom,
                                  float* __restrict__ out) {
    __shared__ float sh[BLOCK_THREADS];
    float s = 0.0f;
    for (int i = threadIdx.x; i < num_partials; i += blockDim.x) s += partials[i];
    sh[threadIdx.x] = s;
    __syncthreads();
    for (int off = blockDim.x >> 1; off > 0; off >>= 1) {
        if ((int)threadIdx.x < off) sh[threadIdx.x] += sh[threadIdx.x + off];
        __syncthreads();
    }
    if (threadIdx.x == 0) {
        float total = sh[0];
        const int tail_start = (n >> 7) << 7;
        for (int i = tail_start; i < n; ++i) {
            float fx = final_out[2 * i];
            float fy = final_out[2 * i + 1];
            float p  = (point_victor[i] == 0) ? fx : fy;
            float d  = fx - 0.5f;
            total += (__builtin_amdgcn_logf(p) * LN2F - d * d)
                     * ((float)i * inv_denom);
        }
        out[0] = -total;
    }
}

extern "C" void kernel_launch(void* const* d_in, const int* in_sizes, int n_in,
                              void* d_out, int out_size, void* d_ws, size_t ws_size,
                              hipStream_t stream) {
    const float* final_out    = (const float*)d_in[0];   // (N, 2) f32
    const int*   point_victor = (const int*)d_in[1];     // (N,)  i32
    float*       out          = (float*)d_out;           // scalar f32
    float*       partials     = (float*)d_ws;            // NUM_BLOCKS floats

    const int n = in_sizes[1];

    // Match the reference: denom computed in f32 (n and n+1 both cast to f32).
    const float denom     = (float)n * (float)(n + 1) * 0.5f;
    const float inv_denom = 1.0f / denom;

    loss_partial_kernel<<<NUM_BLOCKS, BLOCK_THREADS, 0, stream>>>(
        final_out, point_victor, partials, n, inv_denom);
    loss_final_kernel<<<1, BLOCK_THREADS, 0, stream>>>(
        partials, NUM_BLOCKS, final_out, point_victor, n, inv_denom, out);
}